// TGAM_75239237092004
// MI455X (gfx1250) — compile-verified
//
#include <hip/hip_runtime.h>
#include <hip/hip_bf16.h>
#include <math.h>

typedef __attribute__((ext_vector_type(2))) float v2f;
typedef __attribute__((ext_vector_type(8))) float v8f;

#define EPSF   1e-7f
#define K4PI2  0.40528473456935108577f   // 4 / pi^2

#define TILE    1024
#define ROWS_PB 16
#define THREADS 256

// raw v_rcp_f32 (1 ulp) — all denominators in this kernel are provably > 0
__device__ __forceinline__ float frcp(float x) {
    return __builtin_amdgcn_rcpf(x);
}

// p = exp( ciou(i, j) * conf_i * conf_j ), score bounded in (-2, 1] so exp is safe
__device__ __forceinline__ float ciou_p(
    float ix1, float iy1, float ix2, float iy2,
    float isx, float isy,                 // ix1+ix2, iy1+iy2
    float iarea, float iat, float iconf,
    float4 q0, float4 q1)
{
    // LDS row layout: [conf, x1, y1, x2] [y2, 1.0, area, atan]
    float bconf = q0.x, bx1 = q0.y, by1 = q0.z, bx2 = q0.w;
    float by2 = q1.x, barea = q1.z, bat = q1.w;

    float iw = fminf(ix2, bx2) - fmaxf(ix1, bx1); iw = fmaxf(iw, 0.f);
    float ih = fminf(iy2, by2) - fmaxf(iy1, by1); ih = fmaxf(ih, 0.f);
    float inter = iw * ih;
    float uni   = iarea + barea - inter + EPSF;        // >= EPSF
    float cw = fmaxf(ix2, bx2) - fminf(ix1, bx1);
    float ch = fmaxf(iy2, by2) - fminf(iy1, by1);
    float c2 = cw * cw + ch * ch + EPSF;               // >= EPSF
    // fuse 1/uni and 1/c2 into a single quarter-rate rcp:
    // r = 1/(uni*c2); iou = inter*c2*r; rho2/c2 = rho2*uni*r
    float r   = frcp(uni * c2);                        // uni*c2 <= ~1e10, safe
    float iou = inter * c2 * r;
    float dx = (bx1 + bx2) - isx;
    float dy = (by1 + by2) - isy;
    float rho2 = 0.25f * (dx * dx + dy * dy);
    float dat  = bat - iat;
    float vv   = K4PI2 * dat * dat;
    // vv - iou + 1 + eps > 0 since iou <= 1, vv >= 0
    float alpha = vv * frcp(vv - iou + (1.f + EPSF));
    float ciou  = iou - (rho2 * uni * r + vv * alpha);
    return __expf(ciou * iconf * bconf);
}

__global__ __launch_bounds__(THREADS) void ciou_attn_kernel(
    const float* __restrict__ x, const float* __restrict__ gamma,
    float* __restrict__ out, int N)
{
    __shared__ __align__(16) float tileF[TILE * 8];      // 32 KB j-feature tile
    __shared__ float part[ROWS_PB * 6 * 8];              // per-wave partials
    __shared__ float fin[ROWS_PB * 6];

    const int tid  = threadIdx.x;
    const int lane = tid & 31;
    // wave index is wave-uniform: force it into an SGPR so the j-loop is scalar
    // (keeps EXEC all-ones around v_wmma, removes exec-mask loop bookkeeping)
    const int wave = __builtin_amdgcn_readfirstlane(tid >> 5);
    const int mrow = lane & 15;          // A-matrix row M
    const int nsel = lane & 7;           // B-matrix column (N & 7)
    const int i_base = blockIdx.x * ROWS_PB;

    // Per-lane row-i features (lanes 16..31 duplicate 0..15 — needed by A layout)
    const int i = i_base + mrow;
    const float* xi = x + i * 5;
    const float iconf = xi[0];
    const float ix1 = xi[1], iy1 = xi[2], ix2 = xi[3], iy2 = xi[4];
    const float isx = ix1 + ix2, isy = iy1 + iy2;
    const float w1 = ix2 - ix1, h1 = iy2 - iy1;
    const float iarea = w1 * h1;
    const float iat   = atanf(w1 / (h1 + EPSF));

    v8f c0 = {0.f, 0.f, 0.f, 0.f, 0.f, 0.f, 0.f, 0.f};
    v8f c1 = {0.f, 0.f, 0.f, 0.f, 0.f, 0.f, 0.f, 0.f};

    const int koff = (lane < 16) ? 0 : 2;    // A/B K-pair offset for this lane
    const int nTiles = N / TILE;
    for (int t = 0; t < nTiles; ++t) {
        // Cooperative precompute of TILE j-rows into LDS:
        // [conf, x1, y1, x2, y2, 1.0, area, atan]  (atan computed once per box!)
        for (int r = tid; r < TILE; r += THREADS) {
            const float* xj = x + (t * TILE + r) * 5;
            float cf = xj[0], bx1 = xj[1], by1 = xj[2], bx2 = xj[3], by2 = xj[4];
            float w = bx2 - bx1, h = by2 - by1;
            float* row = &tileF[r * 8];
            row[0] = cf;  row[1] = bx1; row[2] = by1; row[3] = bx2;
            row[4] = by2; row[5] = 1.0f; row[6] = w * h;
            row[7] = atanf(w / (h + EPSF));
        }
        __syncthreads();

        // Each wave owns a 128-row j-slice; 8 j's (2 WMMAs, independent
        // accumulators) per loop step so exp->wmma latency is hidden.
        const int jb0 = wave * (TILE / 8);
        const int jb1 = jb0 + (TILE / 8);
        for (int jb = jb0; jb < jb1; jb += 8) {
            // ---- step 0: j's [jb, jb+4) -> c0 ----
            {
                const int jA = jb + koff;
                const float4* rowp = (const float4*)&tileF[jA * 8];
                float4 q0 = rowp[0], q1 = rowp[1];
                float4 r0 = rowp[2], r1 = rowp[3];
                v2f a, b;
                a.x = ciou_p(ix1, iy1, ix2, iy2, isx, isy, iarea, iat, iconf, q0, q1);
                a.y = ciou_p(ix1, iy1, ix2, iy2, isx, isy, iarea, iat, iconf, r0, r1);
                b.x = tileF[jA * 8 + nsel];
                b.y = tileF[jA * 8 + 8 + nsel];
                c0 = __builtin_amdgcn_wmma_f32_16x16x4_f32(
                         false, a, false, b, (short)0, c0, false, false);
            }
            // ---- step 1: j's [jb+4, jb+8) -> c1 ----
            {
                const int jA = jb + 4 + koff;
                const float4* rowp = (const float4*)&tileF[jA * 8];
                float4 q0 = rowp[0], q1 = rowp[1];
                float4 r0 = rowp[2], r1 = rowp[3];
                v2f a, b;
                a.x = ciou_p(ix1, iy1, ix2, iy2, isx, isy, iarea, iat, iconf, q0, q1);
                a.y = ciou_p(ix1, iy1, ix2, iy2, isx, isy, iarea, iat, iconf, r0, r1);
                b.x = tileF[jA * 8 + nsel];
                b.y = tileF[jA * 8 + 8 + nsel];
                c1 = __builtin_amdgcn_wmma_f32_16x16x4_f32(
                         false, a, false, b, (short)0, c1, false, false);
            }
        }
        __syncthreads();
    }

    // merge the two accumulators
    v8f c;
#pragma unroll
    for (int v = 0; v < 8; ++v) c[v] = c0[v] + c1[v];

    // C/D layout: VGPR v, lane l -> M = v + 8*(l/16), N = l%16
    const int Ncol = lane & 15;
    if (Ncol < 6) {
        const int mhi = (lane >> 4) * 8;
#pragma unroll
        for (int v = 0; v < 8; ++v) {
            const int M = v + mhi;
            part[(M * 6 + Ncol) * 8 + wave] = c[v];
        }
    }
    __syncthreads();

    if (tid < ROWS_PB * 6) {
        float s = part[tid * 8];
#pragma unroll
        for (int w = 1; w < 8; ++w) s += part[tid * 8 + w];
        fin[tid] = s;
    }
    __syncthreads();

    if (tid < ROWS_PB * 5) {
        const int row = tid / 5, cc = tid % 5;
        const int ii  = i_base + row;
        const float Z   = fin[row * 6 + 5];      // ones-column = softmax denom
        const float acc = fin[row * 6 + cc];
        const float xp  = x[ii * 5 + cc] * gamma[0] + acc / Z;
        out[ii * 5 + cc] = 1.f / (1.f + __expf(-xp));
    }
}

extern "C" void kernel_launch(void* const* d_in, const int* in_sizes, int n_in,
                              void* d_out, int out_size, void* d_ws, size_t ws_size,
                              hipStream_t stream) {
    const float* x = (const float*)d_in[0];
    const float* g = (const float*)d_in[1];
    float* out = (float*)d_out;
    const int N = in_sizes[0] / 5;               // 4096
    dim3 grid(N / ROWS_PB);                      // 256 blocks x 256 threads (8 waves)
    ciou_attn_kernel<<<grid, THREADS, 0, stream>>>(x, g, out, N);
}